// GCNClassifier_88828513616187
// MI455X (gfx1250) — compile-verified
//
#include <hip/hip_runtime.h>

#define N_NODES 100000
#define N_EDGES 3200000
#define IN_CH   128
#define HID     96
#define HID2    48
#define NCLS    3

typedef __attribute__((ext_vector_type(2)))  float  v2f;
typedef __attribute__((ext_vector_type(8)))  float  v8f;

// Hardware fp32 atomic (global_atomic_add_f32), never a CAS loop.
static __device__ inline void atom_add_f32(float* p, float v) {
    unsafeAtomicAdd(p, v);
}

// ---------------------------------------------------------------- utilities

__global__ __launch_bounds__(256) void init_deg(float* __restrict__ deg) {
    int i = blockIdx.x * blockDim.x + threadIdx.x;
    if (i < N_NODES) deg[i] = 1.0f;   // self-loop weight
}

__global__ __launch_bounds__(256) void accum_deg(const int* __restrict__ dst,
                                                 const float* __restrict__ ew,
                                                 float* __restrict__ deg) {
    int e = blockIdx.x * blockDim.x + threadIdx.x;
    if (e < N_EDGES) atom_add_f32(deg + dst[e], ew[e]);
}

__global__ __launch_bounds__(256) void make_dinv(float* __restrict__ deg) {
    int i = blockIdx.x * blockDim.x + threadIdx.x;
    if (i < N_NODES) deg[i] = rsqrtf(deg[i]);   // deg >= 1 always
}

__global__ __launch_bounds__(256) void make_norm(const int* __restrict__ src,
                                                 const int* __restrict__ dst,
                                                 const float* __restrict__ ew,
                                                 const float* __restrict__ dinv,
                                                 float* __restrict__ norm) {
    int e = blockIdx.x * blockDim.x + threadIdx.x;
    if (e < N_EDGES) norm[e] = dinv[src[e]] * ew[e] * dinv[dst[e]];
}

// ---------------------------------------------------------------- WMMA GEMM
// H[N x COLS] = X[N x K] @ W[K x COLS], fp32, one 16x16 tile per wave.
// A layout (ISA 16x4 f32): lane = M, lane-half picks K{0,1}/{2,3} pair.
// C/D layout: VGPR r -> M = r (+8 for upper lane half), lane&15 = N.

template<int K, int COLS>
__global__ __launch_bounds__(256) void gemm_wmma(const float* __restrict__ X,
                                                 const float* __restrict__ W,
                                                 float* __restrict__ H) {
    const int lane = threadIdx.x & 31;
    const int wave = threadIdx.x >> 5;
    const int tm = blockIdx.x * 8 + wave;          // row tile
    const int tn = blockIdx.y;                     // col tile
    if (tm * 16 >= N_NODES) return;                // wave-uniform guard
    const int m0 = tm * 16, n0 = tn * 16;
    const int row   = m0 + (lane & 15);            // A: lane = M
    const int col   = n0 + (lane & 15);            // B/C/D: lane = N
    const int khalf = (lane >> 4) << 1;            // 0 or 2
    const float* xrow = X + (size_t)row * K;
    v8f c = {};
    for (int kk = 0; kk < K; kk += 4) {
        v2f a = *(const v2f*)(xrow + kk + khalf);  // K even -> 8B aligned
        v2f b;
        b.x = W[(size_t)(kk + khalf)     * COLS + col];
        b.y = W[(size_t)(kk + khalf + 1) * COLS + col];
        c = __builtin_amdgcn_wmma_f32_16x16x4_f32(false, a, false, b,
                                                  (short)0, c, false, false);
    }
    const int rbase = (lane >> 4) << 3;
    float* hrow = H + (size_t)(m0 + rbase) * COLS + col;
    #pragma unroll
    for (int r = 0; r < 8; ++r) hrow[(size_t)r * COLS] = c[r];
}

// ---------------------------------------------------------- message passing

// out[i] = dinv[i]^2 * H[i]   (self-loop contribution seeds the accumulator)
template<int C>
__global__ __launch_bounds__(256) void selfloop_init(const float* __restrict__ H,
                                                     const float* __restrict__ dinv,
                                                     float* __restrict__ out) {
    const int C4 = C / 4;
    int idx = blockIdx.x * blockDim.x + threadIdx.x;
    if (idx >= N_NODES * C4) return;
    int i = idx / C4, q = idx % C4;
    float di = dinv[i], w = di * di;
    float4 hv = *(const float4*)(H + (size_t)i * C + q * 4);
    float4 o = { w * hv.x, w * hv.y, w * hv.z, w * hv.w };
    *(float4*)(out + (size_t)i * C + q * 4) = o;
}

// out[dst] += norm * H[src]; thread = (edge, channel-quad); L2-resident atomics
template<int C, int EPB>
__global__ __launch_bounds__(256) void scatter_edges(const float* __restrict__ H,
                                                     const int* __restrict__ src,
                                                     const int* __restrict__ dst,
                                                     const float* __restrict__ norm,
                                                     float* __restrict__ out) {
    const int q = threadIdx.x;                     // 0 .. C/4-1
    const int e = blockIdx.x * EPB + threadIdx.y;
    if (e >= N_EDGES) return;
    const int s = src[e], d = dst[e];
    const float nw = norm[e];
    const float4 hv = *(const float4*)(H + (size_t)s * C + q * 4);
    float* o = out + (size_t)d * C + q * 4;
    atom_add_f32(o + 0, nw * hv.x);
    atom_add_f32(o + 1, nw * hv.y);
    atom_add_f32(o + 2, nw * hv.z);
    atom_add_f32(o + 3, nw * hv.w);
}

template<int C>
__global__ __launch_bounds__(256) void bias_relu(float* __restrict__ out,
                                                 const float* __restrict__ b) {
    const int C4 = C / 4;
    int idx = blockIdx.x * blockDim.x + threadIdx.x;
    if (idx >= N_NODES * C4) return;
    int i = idx / C4, q = idx % C4;
    float4 v  = *(float4*)(out + (size_t)i * C + q * 4);
    float4 bb = *(const float4*)(b + q * 4);
    v.x = fmaxf(v.x + bb.x, 0.0f);
    v.y = fmaxf(v.y + bb.y, 0.0f);
    v.z = fmaxf(v.z + bb.z, 0.0f);
    v.w = fmaxf(v.w + bb.w, 0.0f);
    *(float4*)(out + (size_t)i * C + q * 4) = v;
}

__global__ __launch_bounds__(256) void classifier(const float* __restrict__ Hin,
                                                  const float* __restrict__ Wc,
                                                  const float* __restrict__ bc,
                                                  float* __restrict__ logits) {
    int i = blockIdx.x * blockDim.x + threadIdx.x;
    if (i >= N_NODES) return;
    const float* h = Hin + (size_t)i * HID2;
    float a0 = bc[0], a1 = bc[1], a2 = bc[2];
    #pragma unroll
    for (int k = 0; k < HID2; ++k) {
        float hv = h[k];
        a0 += hv * Wc[k * 3 + 0];
        a1 += hv * Wc[k * 3 + 1];
        a2 += hv * Wc[k * 3 + 2];
    }
    logits[(size_t)i * 3 + 0] = a0;
    logits[(size_t)i * 3 + 1] = a1;
    logits[(size_t)i * 3 + 2] = a2;
}

// ------------------------------------------------------------------- launch

extern "C" void kernel_launch(void* const* d_in, const int* in_sizes, int n_in,
                              void* d_out, int out_size, void* d_ws, size_t ws_size,
                              hipStream_t stream) {
    const float* x  = (const float*)d_in[0];
    const int*   ei = (const int*)  d_in[1];   // [2, E] int32
    const float* ew = (const float*)d_in[2];
    const float* W1 = (const float*)d_in[3];
    const float* b1 = (const float*)d_in[4];
    const float* W2 = (const float*)d_in[5];
    const float* b2 = (const float*)d_in[6];
    const float* Wc = (const float*)d_in[7];
    const float* bc = (const float*)d_in[8];
    float* logits = (float*)d_out;

    const int* src = ei;
    const int* dst = ei + N_EDGES;

    char* ws = (char*)d_ws;
    size_t off = 0;
    auto take = [&](size_t bytes) -> void* {
        void* p = ws + off;
        off += (bytes + 255) & ~(size_t)255;
        return p;
    };
    float* dinv = (float*)take((size_t)N_NODES * 4);
    float* norm = (float*)take((size_t)N_EDGES * 4);
    float* H1   = (float*)take((size_t)N_NODES * HID * 4);
    float* out1 = (float*)take((size_t)N_NODES * HID * 4);
    float* H2   = H1;     // safe: H1 dead after scatter1+relu, kernels serialize
    float* out2 = out1;   // safe: out1 fully consumed by gemm2 before overwrite

    const int TB = 256;
    init_deg <<<(N_NODES + TB - 1) / TB, TB, 0, stream>>>(dinv);
    accum_deg<<<(N_EDGES + TB - 1) / TB, TB, 0, stream>>>(dst, ew, dinv);
    make_dinv<<<(N_NODES + TB - 1) / TB, TB, 0, stream>>>(dinv);
    make_norm<<<(N_EDGES + TB - 1) / TB, TB, 0, stream>>>(src, dst, ew, dinv, norm);

    // ---- layer 1: GCNConv(128 -> 96) + ReLU
    {
        dim3 g((N_NODES / 16 + 7) / 8, HID / 16);
        gemm_wmma<IN_CH, HID><<<g, 256, 0, stream>>>(x, W1, H1);
    }
    {
        int tot = N_NODES * (HID / 4);
        selfloop_init<HID><<<(tot + TB - 1) / TB, TB, 0, stream>>>(H1, dinv, out1);
        scatter_edges<HID, 8><<<(N_EDGES + 7) / 8, dim3(HID / 4, 8), 0, stream>>>(
            H1, src, dst, norm, out1);
        bias_relu<HID><<<(tot + TB - 1) / TB, TB, 0, stream>>>(out1, b1);
    }

    // ---- layer 2: GCNConv(96 -> 48) + ReLU
    {
        dim3 g((N_NODES / 16 + 7) / 8, HID2 / 16);
        gemm_wmma<HID, HID2><<<g, 256, 0, stream>>>(out1, W2, H2);
    }
    {
        int tot = N_NODES * (HID2 / 4);
        selfloop_init<HID2><<<(tot + TB - 1) / TB, TB, 0, stream>>>(H2, dinv, out2);
        scatter_edges<HID2, 16><<<(N_EDGES + 15) / 16, dim3(HID2 / 4, 16), 0, stream>>>(
            H2, src, dst, norm, out2);
        bias_relu<HID2><<<(tot + TB - 1) / TB, TB, 0, stream>>>(out2, b2);
    }

    // ---- classifier: [N,48] @ [48,3] + bc
    classifier<<<(N_NODES + TB - 1) / TB, TB, 0, stream>>>(out2, Wc, bc, logits);
}